// DiagonalSSMBlock_28097676051128
// MI455X (gfx1250) — compile-verified
//
#include <hip/hip_runtime.h>
#include <hip/hip_bf16.h>

typedef float v2f __attribute__((ext_vector_type(2)));
typedef float v8f __attribute__((ext_vector_type(8)));

#define D_MODEL 1024
#define D_STATE 64
#define SEQ_L   4096
#define BATCH   8
#define T_CH    128                 // time chunk length
#define NCH     (SEQ_L / T_CH)      // 32 chunks per batch
#define KC      64                  // K-blocking for Bu GEMM
#define LN_EPS  1e-5f

// ---- workspace layout (float offsets) ----
#define WS_BBR 0
#define WS_BBI (WS_BBR + D_STATE * D_MODEL)                 // b_bar imag
#define WS_ABR (WS_BBI + D_STATE * D_MODEL)                 // a_bar real [64]
#define WS_ABI (WS_ABR + D_STATE)                           // a_bar imag [64]
#define WS_BU  (WS_ABI + D_STATE)                           // Bu interleaved {re,im} [B][L][N][2]
#define WS_C   (WS_BU + 2 * BATCH * SEQ_L * D_STATE)        // carries interleaved {re,im}
// total = 4,358,272 floats ≈ 17.4 MB

// ------------------------------------------------------------------
// Phase 0: a_bar = exp(a), b_bar = ((a_bar-1)/a) * B_mat
// ------------------------------------------------------------------
__global__ void ssm_setup(const float* __restrict__ lnr, const float* __restrict__ imag,
                          const float* __restrict__ Bm, float* __restrict__ ws) {
    int n = blockIdx.x;                     // 0..63 (one block per state)
    float ar = -expf(lnr[n]);
    float ai = imag[n];
    float ea  = expf(ar);
    float abr = ea * cosf(ai);
    float abi = ea * sinf(ai);
    // coef = (a_bar - 1) / a   (complex division)
    float nr = abr - 1.0f, ni = abi;
    float den = ar * ar + ai * ai;
    float cr = (nr * ar + ni * ai) / den;
    float ci = (ni * ar - nr * ai) / den;
    if (threadIdx.x == 0) {
        ws[WS_ABR + n] = abr;
        ws[WS_ABI + n] = abi;
    }
    for (int d = threadIdx.x; d < D_MODEL; d += blockDim.x) {
        float bv = Bm[n * D_MODEL + d];
        ws[WS_BBR + n * D_MODEL + d] = cr * bv;
        ws[WS_BBI + n * D_MODEL + d] = ci * bv;
    }
}

// ------------------------------------------------------------------
// Phase 1: Bu[b,t,n] = u[b,t,:] . b_bar[n,:]  (re & im) via f32 WMMA.
// One block per (batch, chunk): 256 threads = 8 waves, wave w owns the
// 16-row m-tile starting at t0 + 16*w.
// B fragments live in pair-interleaved LDS: {Bt[2j][n], Bt[2j+1][n]}
// adjacent -> single 8B ds_load into an even VGPR pair (no packing movs).
// ------------------------------------------------------------------
__global__ __launch_bounds__(256) void ssm_bu(const float* __restrict__ u,
                                              float* __restrict__ ws) {
    __shared__ float u_lds[T_CH][KC];             // 32 KB
    __shared__ float btpr[KC / 2][D_STATE * 2];   // 16 KB  paired b_bar^T real
    __shared__ float btpi[KC / 2][D_STATE * 2];   // 16 KB  paired b_bar^T imag

    const int blk = blockIdx.x;
    const int b = blk / NCH, c = blk % NCH;
    const int t0 = c * T_CH;
    const int tid = threadIdx.x;
    const int w = tid >> 5, l = tid & 31;
    const int l16 = l & 15, hgrp = l >> 4;

    v8f accR[4], accI[4];
#pragma unroll
    for (int nt = 0; nt < 4; ++nt) { accR[nt] = {}; accI[nt] = {}; }

    const float* bbr = ws + WS_BBR;
    const float* bbi = ws + WS_BBI;

    for (int k0 = 0; k0 < D_MODEL; k0 += KC) {
        // stage u tile: T_CH x KC (8 float4 per thread)
#pragma unroll
        for (int q = 0; q < 8; ++q) {
            int lin = tid + q * 256;                 // 0..2047 float4 slots
            int row = lin >> 4, fc = lin & 15;
            const float4 v = *(const float4*)(u + ((size_t)(b * SEQ_L + t0 + row)) * D_MODEL + k0 + fc * 4);
            *(float4*)&u_lds[row][fc * 4] = v;
        }
        // stage b_bar^T tiles in pair-interleaved layout (re+im)
#pragma unroll
        for (int p = 0; p < 4; ++p) {
            int lin = tid + p * 256;                 // 0..1023 float4 slots
            int n = lin >> 4, kf = lin & 15;         // k = kf*4 .. kf*4+3
            float4 vr = *(const float4*)(bbr + (size_t)n * D_MODEL + k0 + kf * 4);
            float4 vi = *(const float4*)(bbi + (size_t)n * D_MODEL + k0 + kf * 4);
            v2f t;
            t.x = vr.x; t.y = vr.y; *(v2f*)&btpr[kf * 2 + 0][n * 2] = t;
            t.x = vr.z; t.y = vr.w; *(v2f*)&btpr[kf * 2 + 1][n * 2] = t;
            t.x = vi.x; t.y = vi.y; *(v2f*)&btpi[kf * 2 + 0][n * 2] = t;
            t.x = vi.z; t.y = vi.w; *(v2f*)&btpi[kf * 2 + 1][n * 2] = t;
        }
        // prefetch next K-slab of u while we compute on this one
        if (k0 + KC < D_MODEL) {
            int lin = tid;                           // one line per thread
            int row = lin >> 1, half = lin & 1;
            __builtin_prefetch(u + ((size_t)(b * SEQ_L + t0 + row)) * D_MODEL + (k0 + KC) + half * 32, 0, 3);
        }
        __syncthreads();

#pragma unroll
        for (int ks = 0; ks < KC / 4; ++ks) {
            const int kk = ks * 4 + 2 * hgrp;        // ISA 16x4 A layout
            const int j = kk >> 1;                   // pair row in btp*
            v2f a = *(const v2f*)&u_lds[w * 16 + l16][kk];
#pragma unroll
            for (int nt = 0; nt < 4; ++nt) {
                v2f br_ = *(const v2f*)&btpr[j][(nt * 16 + l16) * 2];
                v2f bi_ = *(const v2f*)&btpi[j][(nt * 16 + l16) * 2];
                accR[nt] = __builtin_amdgcn_wmma_f32_16x16x4_f32(
                    false, a, false, br_, (short)0, accR[nt], false, false);
                accI[nt] = __builtin_amdgcn_wmma_f32_16x16x4_f32(
                    false, a, false, bi_, (short)0, accI[nt], false, false);
            }
        }
        __syncthreads();
    }

    // store Bu interleaved {re,im}: one 8B store per element, lanes 0-15
    // cover 16 consecutive n -> 128B contiguous segments
    float* bu = ws + WS_BU;
#pragma unroll
    for (int nt = 0; nt < 4; ++nt) {
        int n = nt * 16 + l16;
#pragma unroll
        for (int v = 0; v < 8; ++v) {
            int gt = t0 + w * 16 + v + 8 * hgrp;     // ISA C/D layout row
            size_t idx = ((size_t)(b * SEQ_L + gt)) * D_STATE + n;
            v2f pr;
            pr.x = accR[nt][v];
            pr.y = accI[nt][v];
            *(v2f*)(bu + idx * 2) = pr;
        }
    }
}

// ------------------------------------------------------------------
// Phase 2: sequential carry scan (exact reference ordering).
// 8 blocks (one per batch) x 64 lanes (one per state).
// ------------------------------------------------------------------
__global__ void ssm_scan_carry(float* __restrict__ ws) {
    const int b = blockIdx.x;
    const int n = threadIdx.x;
    const float ar = ws[WS_ABR + n], ai = ws[WS_ABI + n];
    const float* bu = ws + WS_BU;
    float* cw = ws + WS_C;
    float hr = 0.f, hi = 0.f;
    for (int t = 0; t < SEQ_L; ++t) {
        if ((t & (T_CH - 1)) == 0) {
            int c = t >> 7;  // t / T_CH
            v2f pr; pr.x = hr; pr.y = hi;
            *(v2f*)(cw + ((size_t)(b * NCH + c) * D_STATE + n) * 2) = pr;
        }
        size_t idx = ((size_t)(b * SEQ_L + t)) * D_STATE + n;
        v2f bv = *(const v2f*)(bu + idx * 2);
        float nhr = hr * ar - hi * ai + bv.x;
        float nhi = hr * ai + hi * ar + bv.y;
        hr = nhr; hi = nhi;
    }
}

// ------------------------------------------------------------------
// Phase 3: per-chunk re-scan with carry, readout GEMM h_re @ C^T via
// f32 WMMA, fused residual + D*u + LayerNorm. One block per chunk.
// ------------------------------------------------------------------
__global__ __launch_bounds__(256) void ssm_out(const float* __restrict__ u,
                                               const float* __restrict__ Cm,
                                               const float* __restrict__ Dv,
                                               const float* __restrict__ gamma,
                                               const float* __restrict__ beta,
                                               const float* __restrict__ ws,
                                               float* __restrict__ out) {
    __shared__ float hre[T_CH][D_STATE];   // 32 KB

    const int blk = blockIdx.x;
    const int b = blk / NCH, c = blk % NCH;
    const int t0 = c * T_CH;
    const int tid = threadIdx.x;

    if (tid < D_STATE) {   // waves 0-1 fully active: no intra-wave divergence
        const int n = tid;
        const float ar = ws[WS_ABR + n], ai = ws[WS_ABI + n];
        v2f cv = *(const v2f*)(ws + WS_C + ((size_t)(b * NCH + c) * D_STATE + n) * 2);
        float hr = cv.x, hi = cv.y;
        const float* bu = ws + WS_BU;
        for (int t = 0; t < T_CH; ++t) {
            size_t idx = ((size_t)(b * SEQ_L + t0 + t)) * D_STATE + n;
            v2f bv = *(const v2f*)(bu + idx * 2);
            float nhr = hr * ar - hi * ai + bv.x;
            float nhi = hr * ai + hi * ar + bv.y;
            hr = nhr; hi = nhi;
            hre[t][n] = hr;
        }
    }
    __syncthreads();

    const int w = tid >> 5, l = tid & 31;
    const int l16 = l & 15, hgrp = l >> 4;

    float rs[8], rq[8];
#pragma unroll
    for (int v = 0; v < 8; ++v) { rs[v] = 0.f; rq[v] = 0.f; }

    for (int dt = 0; dt < D_MODEL / 16; ++dt) {
        v8f acc = {};
        const int dcol = dt * 16 + l16;
        const float* crow = Cm + (size_t)dcol * D_STATE;   // C_mat[dcol][k], contiguous in k
#pragma unroll
        for (int ks = 0; ks < D_STATE / 4; ++ks) {
            const int kk = ks * 4 + 2 * hgrp;
            v2f a  = *(const v2f*)&hre[w * 16 + l16][kk];  // 8B LDS load
            v2f bf = *(const v2f*)(crow + kk);             // 8B global load (L2-hot)
            acc = __builtin_amdgcn_wmma_f32_16x16x4_f32(
                false, a, false, bf, (short)0, acc, false, false);
        }
        const float dscale = 1.0f + Dv[dcol];
#pragma unroll
        for (int v = 0; v < 8; ++v) {
            int gt = t0 + w * 16 + v + 8 * hgrp;
            size_t idx = ((size_t)(b * SEQ_L + gt)) * D_MODEL + dcol;
            float x = acc[v] + u[idx] * dscale;    // u + y + D*u
            out[idx] = x;
            rs[v] += x;
            rq[v] += x * x;
        }
    }

    // reduce row sums across the 16 lanes covering each row (wave32 shuffles)
#pragma unroll
    for (int v = 0; v < 8; ++v) {
        float s = rs[v], q = rq[v];
#pragma unroll
        for (int m = 1; m < 16; m <<= 1) {
            s += __shfl_xor(s, m, 32);
            q += __shfl_xor(q, m, 32);
        }
        rs[v] = s; rq[v] = q;
    }

    // normalize (re-read rows just written: same-lane same-address, L2-hot)
    for (int dt = 0; dt < D_MODEL / 16; ++dt) {
        const int dcol = dt * 16 + l16;
        const float g = gamma[dcol];
        const float be = beta[dcol];
#pragma unroll
        for (int v = 0; v < 8; ++v) {
            float mu = rs[v] * (1.0f / D_MODEL);
            float var = rq[v] * (1.0f / D_MODEL) - mu * mu;
            float inv = rsqrtf(var + LN_EPS);
            int gt = t0 + w * 16 + v + 8 * hgrp;
            size_t idx = ((size_t)(b * SEQ_L + gt)) * D_MODEL + dcol;
            float x = out[idx];
            out[idx] = (x - mu) * inv * g + be;
        }
    }
}

// ------------------------------------------------------------------
extern "C" void kernel_launch(void* const* d_in, const int* in_sizes, int n_in,
                              void* d_out, int out_size, void* d_ws, size_t ws_size,
                              hipStream_t stream) {
    (void)in_sizes; (void)n_in; (void)out_size; (void)ws_size;
    const float* u     = (const float*)d_in[0];
    const float* lnr   = (const float*)d_in[1];
    const float* imag  = (const float*)d_in[2];
    const float* Bm    = (const float*)d_in[3];
    const float* Cm    = (const float*)d_in[4];
    const float* Dv    = (const float*)d_in[5];
    const float* gam   = (const float*)d_in[6];
    const float* bet   = (const float*)d_in[7];
    float* out = (float*)d_out;
    float* ws  = (float*)d_ws;

    ssm_setup<<<dim3(D_STATE), dim3(256), 0, stream>>>(lnr, imag, Bm, ws);
    ssm_bu<<<dim3(BATCH * NCH), dim3(256), 0, stream>>>(u, ws);
    ssm_scan_carry<<<dim3(BATCH), dim3(D_STATE), 0, stream>>>(ws);
    ssm_out<<<dim3(BATCH * NCH), dim3(256), 0, stream>>>(u, Cm, Dv, gam, bet, ws, out);
}